// AttentionHead_5351529251283
// MI455X (gfx1250) — compile-verified
//
#include <hip/hip_runtime.h>
#include <math.h>

// Problem constants (from reference)
constexpr int B  = 4;
constexpr int NQ = 256;
constexpr int NK = 512;
constexpr int H  = 256;

typedef __attribute__((ext_vector_type(2))) float v2f;
typedef __attribute__((ext_vector_type(8))) float v8f;

__device__ __forceinline__ float fast_tanh(float x) {
#if __has_builtin(__builtin_amdgcn_tanhf)
    return __builtin_amdgcn_tanhf(x);
#else
    return tanhf(x);
#endif
}

// ---------------------------------------------------------------------------
// Kernel 1: Out[b, r, o] = sum_h X[b, r, h] * W[o, h]   (i.e. X @ W^T)
// One wave (32 threads) per 16x16 output tile, fp32 WMMA 16x16x4.
//
// A (16x4, M x K) fp32 layout: lane L(0..15): M=L, VGPR0=K0, VGPR1=K1;
//                              lane 16+L:     M=L, VGPR0=K2, VGPR1=K3.
// B (4x16, K x N) fp32 layout mirrors A with N on lanes.
// D (16x16) f32:  lane L(0..15): N=L, VGPR r -> M=r; lane 16+L -> M=8+r.
// ---------------------------------------------------------------------------
__global__ __launch_bounds__(32)
void gemm_xwT_wmma(const float* __restrict__ X, const float* __restrict__ W,
                   float* __restrict__ Out, int rows) {
    const int tilesO = H / 16;                   // 16 col tiles
    const int tilesR = rows / 16;                // row tiles
    int tile = blockIdx.x;
    int o0   = (tile % tilesO) * 16;
    int r0   = ((tile / tilesO) % tilesR) * 16;
    int b    = tile / (tilesO * tilesR);

    int lane = threadIdx.x;
    int lo   = lane & 15;        // M (for A) / N (for B,D)
    int hi   = lane >> 4;        // selects K pair {0,1} vs {2,3}

    const float* Arow = X + ((size_t)b * rows + r0 + lo) * H + 2 * hi;
    const float* Brow = W + (size_t)(o0 + lo) * H + 2 * hi;

    v8f acc = {};
#pragma unroll 8
    for (int h = 0; h < H; h += 4) {
        v2f a = *(const v2f*)(Arow + h);
        v2f w = *(const v2f*)(Brow + h);
        acc = __builtin_amdgcn_wmma_f32_16x16x4_f32(
            /*neg_a=*/false, a, /*neg_b=*/false, w,
            /*c_mod=*/(short)0, acc, /*reuse_a=*/false, /*reuse_b=*/false);
    }

    float* Obase = Out + ((size_t)b * rows + r0 + 8 * hi) * H + o0 + lo;
#pragma unroll
    for (int r = 0; r < 8; ++r)
        Obase[(size_t)r * H] = acc[r];
}

// ---------------------------------------------------------------------------
// Kernel 2: prod[b,q,k] = sum_h tanh(kt[b,k,h] + qt[b,q,h]) * v[h]
// Block = (32 k) x (16 q) threads; tiles staged in LDS with stride-257
// padding so the per-lane k-stride hits 32 distinct banks.
// ---------------------------------------------------------------------------
constexpr int KT   = 32;
constexpr int QT   = 16;
constexpr int PAD  = H + 1;      // 257-float padded row stride

__global__ __launch_bounds__(KT * QT)
void tanh_dot_kernel(const float* __restrict__ kt, const float* __restrict__ qt,
                     const float* __restrict__ v, float* __restrict__ prod) {
    __shared__ float ktS[KT * PAD];
    __shared__ float qtS[QT * PAD];
    __shared__ float vS[H];

    int k0 = blockIdx.x * KT;
    int q0 = blockIdx.y * QT;
    int b  = blockIdx.z;

    int tid = threadIdx.y * KT + threadIdx.x;    // 0..511

    // Stage kt tile (KT x H), qt tile (QT x H), v (H) into LDS.
    const float* ktG = kt + ((size_t)b * NK + k0) * H;
    for (int i = tid; i < KT * H; i += KT * QT)
        ktS[(i >> 8) * PAD + (i & (H - 1))] = ktG[i];

    const float* qtG = qt + ((size_t)b * NQ + q0) * H;
    for (int i = tid; i < QT * H; i += KT * QT)
        qtS[(i >> 8) * PAD + (i & (H - 1))] = qtG[i];

    for (int i = tid; i < H; i += KT * QT)
        vS[i] = v[i];

    __syncthreads();

    const float* kRow = &ktS[threadIdx.x * PAD];
    const float* qRow = &qtS[threadIdx.y * PAD];

    float acc = 0.0f;
#pragma unroll 8
    for (int h = 0; h < H; ++h)
        acc += fast_tanh(kRow[h] + qRow[h]) * vS[h];

    prod[(((size_t)b * NQ + q0 + threadIdx.y) * NK) + k0 + threadIdx.x] = acc;
}

// ---------------------------------------------------------------------------
// Kernel 3: log_softmax along the NK axis. One block per (b,q) row.
// ---------------------------------------------------------------------------
__global__ __launch_bounds__(NK)
void logsoftmax_kernel(const float* __restrict__ prod, float* __restrict__ out) {
    __shared__ float red[NK];
    int row = blockIdx.x;                        // b*NQ + q
    int t   = threadIdx.x;

    const float* x = prod + (size_t)row * NK;
    float val = x[t];

    red[t] = val;
    __syncthreads();
#pragma unroll
    for (int s = NK / 2; s > 0; s >>= 1) {
        if (t < s) red[t] = fmaxf(red[t], red[t + s]);
        __syncthreads();
    }
    float m = red[0];
    __syncthreads();

    float e = __expf(val - m);
    red[t] = e;
    __syncthreads();
#pragma unroll
    for (int s = NK / 2; s > 0; s >>= 1) {
        if (t < s) red[t] += red[t + s];
        __syncthreads();
    }
    float lse = m + __logf(red[0]);

    out[(size_t)row * NK + t] = val - lse;
}

// ---------------------------------------------------------------------------
extern "C" void kernel_launch(void* const* d_in, const int* in_sizes, int n_in,
                              void* d_out, int out_size, void* d_ws, size_t ws_size,
                              hipStream_t stream) {
    const float* x_query = (const float*)d_in[0];   // (B, NQ, H)
    const float* x_key   = (const float*)d_in[1];   // (B, NK, H)
    const float* w1      = (const float*)d_in[2];   // (H, H)  (out,in)
    const float* w2      = (const float*)d_in[3];   // (H, H)
    const float* v       = (const float*)d_in[4];   // (1, H)
    float* out = (float*)d_out;                     // (B, NQ, NK)

    float* kt   = (float*)d_ws;                     // B*NK*H
    float* qt   = kt + (size_t)B * NK * H;          // B*NQ*H
    float* prod = qt + (size_t)B * NQ * H;          // B*NQ*NK

    // kt = x_key @ w1^T
    gemm_xwT_wmma<<<B * (NK / 16) * (H / 16), 32, 0, stream>>>(x_key, w1, kt, NK);
    // qt = x_query @ w2^T
    gemm_xwT_wmma<<<B * (NQ / 16) * (H / 16), 32, 0, stream>>>(x_query, w2, qt, NQ);

    // prod = sum_h tanh(kt + qt) * v
    tanh_dot_kernel<<<dim3(NK / KT, NQ / QT, B), dim3(KT, QT), 0, stream>>>(kt, qt, v, prod);

    // out = log_softmax(prod, axis=-1)
    logsoftmax_kernel<<<B * NQ, NK, 0, stream>>>(prod, out);
}